// SPDSRUcell_39402029974126
// MI455X (gfx1250) — compile-verified
//
#include <hip/hip_runtime.h>

typedef __attribute__((ext_vector_type(2))) float v2f;
typedef __attribute__((ext_vector_type(8))) float v8f;

#define EPS_F 1e-10f
#define WP_OFF 0
#define WY_OFF 8192
#define SC_OFF 12288   // [0..4]=wr, [5..9]=c, [10]=d

__device__ __forceinline__ float4 f4_fma(float s, const float4& a, const float4& b) {
  return make_float4(fmaf(s,a.x,b.x), fmaf(s,a.y,b.y), fmaf(s,a.z,b.z), fmaf(s,a.w,b.w));
}
__device__ __forceinline__ float4 f4_mix(float p, const float4& a, float q, const float4& b) {
  return make_float4(p*a.x+q*b.x, p*a.y+q*b.y, p*a.z+q*b.z, p*a.w+q*b.w);
}
__device__ __forceinline__ float4 f4_scale(float s, const float4& a) {
  return make_float4(s*a.x, s*a.y, s*a.z, s*a.w);
}

// Fragment-ordered weight index: value W[k][n] goes to (k>>2)*256 + n*4 + (k&3).
// Lane reads its B fragment {W[4s+2h][c], W[4s+2h+1][c]} as one 8-byte LDS load.
__device__ __forceinline__ int wfrag_idx(int k, int n) {
  return ((k >> 2) << 8) + (n << 2) + (k & 3);
}

// ---------------- setup: Cayley transforms + weight assembly ----------------
__device__ void cayley8(const float* __restrict__ b, float* __restrict__ q) {
  // L per reference: row i+1, cols 0..i get b[i..2i]
  float L[8][8];
  for (int i = 0; i < 8; ++i) for (int j = 0; j < 8; ++j) L[i][j] = 0.f;
  for (int i = 0; i < 7; ++i) for (int c = 0; c <= i; ++c) L[i+1][c] = b[i + c];
  // M = [I-A | I+A], A = L - L^T ; Gauss-Jordan with partial pivoting
  float M[8][16];
  for (int i = 0; i < 8; ++i) for (int j = 0; j < 8; ++j) {
    float a = L[i][j] - L[j][i];
    float id = (i == j) ? 1.f : 0.f;
    M[i][j]     = id - a;
    M[i][8 + j] = id + a;
  }
  for (int col = 0; col < 8; ++col) {
    int piv = col; float mx = fabsf(M[col][col]);
    for (int r = col + 1; r < 8; ++r) { float v = fabsf(M[r][col]); if (v > mx) { mx = v; piv = r; } }
    if (piv != col) for (int j = 0; j < 16; ++j) { float t = M[col][j]; M[col][j] = M[piv][j]; M[piv][j] = t; }
    float inv = 1.f / M[col][col];
    for (int j = 0; j < 16; ++j) M[col][j] *= inv;
    for (int r = 0; r < 8; ++r) {
      if (r == col) continue;
      float f = M[r][col];
      for (int j = 0; j < 16; ++j) M[r][j] = fmaf(-f, M[col][j], M[r][j]);
    }
  }
  for (int i = 0; i < 8; ++i) for (int j = 0; j < 8; ++j) q[i*8 + j] = M[i][8 + j];
}

__global__ __launch_bounds__(64) void spdsru_setup(
    const float* __restrict__ kr, const float* __restrict__ kt,
    const float* __restrict__ kphi, const float* __restrict__ ks,
    const float* __restrict__ br, const float* __restrict__ bt,
    const float* __restrict__ by, float* __restrict__ wsf)
{
  __shared__ float qs[3*64];   // Qr, Qt, Qy
  __shared__ float a1[64];     // Qt*Qr
  __shared__ float sc[2];      // 1-t, t
  const int tid = threadIdx.x;
  if      (tid == 0) cayley8(br, &qs[0]);
  else if (tid == 1) cayley8(bt, &qs[64]);
  else if (tid == 2) cayley8(by, &qs[128]);
  if (tid == 3) {
    const float alpha[5] = {0.01f, 0.25f, 0.5f, 0.9f, 0.99f};
    float kt2 = kt[0]*kt[0], kp2 = kphi[0]*kphi[0];
    float t = kt2 / (kt2 + kp2 + EPS_F);
    float sr = 0.f, ss = 0.f;
    for (int a = 0; a < 5; ++a) { sr += kr[a]*kr[a]; ss += ks[a]*ks[a]; }
    float d = 0.f;
    for (int a = 0; a < 5; ++a) {
      float wra = kr[a]*kr[a] / (sr + EPS_F);
      float wsa = ks[a]*ks[a] / (ss + EPS_F);
      wsf[SC_OFF + a]     = wra;
      wsf[SC_OFF + 5 + a] = wsa * (1.f - alpha[a]);
      d += wsa * alpha[a];
    }
    wsf[SC_OFF + 10] = d;
    sc[0] = 1.f - t; sc[1] = t;
  }
  __syncthreads();
  { // A1 = Qt * Qr
    int i = tid >> 3, j = tid & 7;
    float s = 0.f;
    for (int m = 0; m < 8; ++m) s = fmaf(qs[64 + i*8 + m], qs[m*8 + j], s);
    a1[i*8 + j] = s;
  }
  __syncthreads();
  const float one_t = sc[0], tt = sc[1];
  // Wp[k][n]: k<64 -> (1-t)*Qt[n1,k1]*Qt[n2,k2] ; k>=64 -> t*A1[n1,k1]*A1[n2,k2]
  for (int idx = tid; idx < 128*64; idx += 64) {
    int k = idx >> 6, n = idx & 63, n1 = n >> 3, n2 = n & 7;
    float v;
    if (k < 64) { int k1 = k >> 3, k2 = k & 7; v = one_t * qs[64 + n1*8 + k1] * qs[64 + n2*8 + k2]; }
    else        { int kk = k - 64; int k1 = kk >> 3, k2 = kk & 7; v = tt * a1[n1*8 + k1] * a1[n2*8 + k2]; }
    wsf[WP_OFF + wfrag_idx(k, n)] = v;
  }
  // Wy[k][n] = Qy[n1,k1]*Qy[n2,k2]
  for (int idx = tid; idx < 64*64; idx += 64) {
    int k = idx >> 6, n = idx & 63, n1 = n >> 3, n2 = n & 7, k1 = k >> 3, k2 = k & 7;
    wsf[WY_OFF + wfrag_idx(k, n)] = qs[128 + n1*8 + k1] * qs[128 + n2*8 + k2];
  }
}

// ---------------- main streaming kernel ----------------
#define WMMA4(A, B, C) __builtin_amdgcn_wmma_f32_16x16x4_f32(false, (A), false, (B), (short)0, (C), false, false)

__global__ __launch_bounds__(256) void spdsru_main(
    const float* __restrict__ x, const float* __restrict__ st,
    const float* __restrict__ wsf, float* __restrict__ oo, float* __restrict__ om,
    int Btot, int numTiles)
{
  __shared__ float wp[128*64];   // GEMM1 weights (Phit), fragment-ordered
  __shared__ float wy[64*64];    // GEMM2 weights (Ot), fragment-ordered
  __shared__ float u2[128*132];  // per-element [x|Y], later S2 (padded rows)
  __shared__ float bp[128*68];   // Phit / Ot staging (padded rows)
  const int tid = threadIdx.x;
  { // weights -> LDS (float4); fragment ordering is a permutation, plain copy works
    const float4* s0 = (const float4*)(wsf + WP_OFF);
    float4* d0 = (float4*)wp;
    for (int i = tid; i < 2048; i += 256) d0[i] = s0[i];
    const float4* s1 = (const float4*)(wsf + WY_OFF);
    float4* d1 = (float4*)wy;
    for (int i = tid; i < 1024; i += 256) d1[i] = s1[i];
  }
  float wr[5], cc[5];
#pragma unroll
  for (int a = 0; a < 5; ++a) { wr[a] = wsf[SC_OFF + a]; cc[a] = wsf[SC_OFF + 5 + a]; }
  const float dd = wsf[SC_OFF + 10];
  const float alpha[5] = {0.01f, 0.25f, 0.5f, 0.9f, 0.99f};

  const int e_loc = tid >> 1;            // local element 0..127 (2 threads / element)
  const int j0    = (tid & 1) * 32;      // component half
  const int lane  = tid & 31;
  const int wv    = tid >> 5;            // wave id, 16 elements each
  const int arow  = (wv << 4) + (lane & 15);        // A-fragment row in u2
  const int kadd  = (lane >> 4) << 1;               // +0 (lanes 0-15) / +2 (lanes 16-31)
  const int bb    = ((lane & 15) << 2) + ((lane >> 4) << 1); // B-fragment lane offset
  const int rbase = (wv << 4) + ((lane >> 4) << 3); // C-tile element row base
  const int cbase = lane & 15;                      // C column within tile
  __syncthreads();

  for (int tile = blockIdx.x; tile < numTiles; tile += gridDim.x) {
    const int e = tile * 128 + e_loc;
    const bool ok = (e < Btot);
    const size_t ec = ok ? (size_t)e : (size_t)(Btot - 1);

    // ---- S1: load x, compute Y = sum wr*M, stage [x|Y] into LDS ----
    {
      float4 xv[8], yv[8];
      const float4* xp = (const float4*)(x + ec*64 + j0);
#pragma unroll
      for (int q = 0; q < 8; ++q) { xv[q] = xp[q]; yv[q] = make_float4(0,0,0,0); }
#pragma unroll
      for (int a = 0; a < 5; ++a) {
        const float4* mp = (const float4*)(st + ec*320 + a*64 + j0);
#pragma unroll
        for (int q = 0; q < 8; ++q) { float4 m = mp[q]; yv[q] = f4_fma(wr[a], m, yv[q]); }
      }
      float* ur = &u2[e_loc * 132];
#pragma unroll
      for (int q = 0; q < 8; ++q) {
        *(float4*)&ur[j0 + 4*q]      = xv[q];
        *(float4*)&ur[64 + j0 + 4*q] = yv[q];
      }
    }
    __syncthreads();

    // ---- GEMM1: Phit[16x64] = u2[16x128] * Wp[128x64], per wave ----
    v8f c0 = {}, c1 = {}, c2 = {}, c3 = {};
#pragma unroll
    for (int s = 0; s < 32; ++s) {
      const int k = 4*s + kadd;
      const float* ua = &u2[arow*132 + k];
      v2f av; av.x = ua[0]; av.y = ua[1];
      const float* wf = &wp[(s << 8) + bb];
      v2f b0 = *(const v2f*)&wf[0];
      v2f b1 = *(const v2f*)&wf[64];
      v2f b2 = *(const v2f*)&wf[128];
      v2f b3 = *(const v2f*)&wf[192];
      c0 = WMMA4(av, b0, c0); c1 = WMMA4(av, b1, c1);
      c2 = WMMA4(av, b2, c2); c3 = WMMA4(av, b3, c3);
    }
#pragma unroll
    for (int kk = 0; kk < 8; ++kk) {
      float* row = &bp[(rbase + kk)*68 + cbase];
      row[0] = c0[kk]; row[16] = c1[kk]; row[32] = c2[kk]; row[48] = c3[kk];
    }
    __syncthreads();

    // ---- S3: Mt = (1-a)M + a*Phit (store), S2 = sum cc*M + d*Phit (stage) ----
    {
      float4 pv[8], s2v[8];
      const float* pr = &bp[e_loc*68 + j0];
#pragma unroll
      for (int q = 0; q < 8; ++q) { pv[q] = *(const float4*)&pr[4*q]; s2v[q] = f4_scale(dd, pv[q]); }
#pragma unroll
      for (int a = 0; a < 5; ++a) {
        const float4* mp = (const float4*)(st + ec*320 + a*64 + j0);  // L2 hit
        float4* op = (float4*)(om + (size_t)e*320 + a*64 + j0);
        const float oma = 1.f - alpha[a], al = alpha[a];
#pragma unroll
        for (int q = 0; q < 8; ++q) {
          float4 m = mp[q];
          float4 r = f4_mix(oma, m, al, pv[q]);
          if (ok) op[q] = r;
          s2v[q] = f4_fma(cc[a], m, s2v[q]);
        }
      }
      float* s2r = &u2[e_loc*132 + j0];
#pragma unroll
      for (int q = 0; q < 8; ++q) *(float4*)&s2r[4*q] = s2v[q];
    }
    __syncthreads();

    // ---- GEMM2: Ot[16x64] = S2[16x64] * Wy[64x64], per wave ----
    v8f d0 = {}, d1 = {}, d2v = {}, d3 = {};
#pragma unroll
    for (int s = 0; s < 16; ++s) {
      const int k = 4*s + kadd;
      const float* ua = &u2[arow*132 + k];
      v2f av; av.x = ua[0]; av.y = ua[1];
      const float* wf = &wy[(s << 8) + bb];
      v2f b0 = *(const v2f*)&wf[0];
      v2f b1 = *(const v2f*)&wf[64];
      v2f b2 = *(const v2f*)&wf[128];
      v2f b3 = *(const v2f*)&wf[192];
      d0 = WMMA4(av, b0, d0); d1 = WMMA4(av, b1, d1);
      d2v = WMMA4(av, b2, d2v); d3 = WMMA4(av, b3, d3);
    }
#pragma unroll
    for (int kk = 0; kk < 8; ++kk) {
      float* row = &bp[(rbase + kk)*68 + cbase];
      row[0] = d0[kk]; row[16] = d1[kk]; row[32] = d2v[kk]; row[48] = d3[kk];
    }
    __syncthreads();

    // ---- final: coalesced store of Ot ----
    if (ok) {
      float4* oor = (float4*)(oo + (size_t)e*64 + j0);
      const float* pr = &bp[e_loc*68 + j0];
#pragma unroll
      for (int q = 0; q < 8; ++q) oor[q] = *(const float4*)&pr[4*q];
    }
    __syncthreads();
  }
}

extern "C" void kernel_launch(void* const* d_in, const int* in_sizes, int n_in,
                              void* d_out, int out_size, void* d_ws, size_t ws_size,
                              hipStream_t stream) {
  const float* x    = (const float*)d_in[0];
  const float* st   = (const float*)d_in[1];
  const float* kr   = (const float*)d_in[2];
  const float* kt   = (const float*)d_in[3];
  const float* kphi = (const float*)d_in[4];
  const float* ks   = (const float*)d_in[5];
  const float* br   = (const float*)d_in[6];
  const float* bt   = (const float*)d_in[7];
  const float* by   = (const float*)d_in[8];
  float* wsf = (float*)d_ws;
  const int B = in_sizes[0] / 64;

  spdsru_setup<<<1, 64, 0, stream>>>(kr, kt, kphi, ks, br, bt, by, wsf);

  const int numTiles = (B + 127) / 128;
  const int grid = numTiles < 2048 ? numTiles : 2048;
  float* oo = (float*)d_out;
  float* om = oo + (size_t)B * 64;
  spdsru_main<<<grid, 256, 0, stream>>>(x, st, wsf, oo, om, B, numTiles);
}